// GAT_53936199303748
// MI455X (gfx1250) — compile-verified
//
#include <hip/hip_runtime.h>
#include <math.h>

#define N_NODES 50000
#define E_EDGES 800000
#define SLOPE 0.2f

typedef __attribute__((ext_vector_type(2))) float v2f;
typedef __attribute__((ext_vector_type(8))) float v8f;

// ---- monotone float <-> uint encoding so float-max == uint-max ----
__device__ __forceinline__ unsigned fenc(float x) {
    unsigned u = __float_as_uint(x);
    return (u & 0x80000000u) ? ~u : (u | 0x80000000u);
}
__device__ __forceinline__ float fdec(unsigned e) {
    return __uint_as_float((e & 0x80000000u) ? (e & 0x7FFFFFFFu) : ~e);
}

// ============================================================
// Dense projection: C[M x NCOLS] = A[M x K] * B[K x NCOLS]
// M = 50000 (multiple of 16). One block = one 16-row tile of A
// (staged in LDS, padded stride) x 8 waves, each wave owns one
// 16x16 output tile, K/4 v_wmma_f32_16x16x4_f32 steps.
// ============================================================
template <int K, int NCOLS>
__launch_bounds__(256)
__global__ void wmma_gemm_kernel(const float* __restrict__ A,
                                 const float* __restrict__ B,
                                 float* __restrict__ C) {
    constexpr int LDA = K + 2;                 // pad: conflict-free b64 LDS reads
    __shared__ float As[16 * LDA];
    const int mt  = blockIdx.x;
    const int tid = threadIdx.x;

    const float* Arow = A + (size_t)mt * 16 * K;
    #pragma unroll
    for (int i = tid; i < 16 * K; i += 256) {
        int r = i / K, c = i & (K - 1);
        As[r * LDA + c] = Arow[i];
    }
    __syncthreads();

    const int wave = tid >> 5;
    const int lane = tid & 31;
    const int n0   = (blockIdx.y * 8 + wave) * 16;
    if (n0 >= NCOLS) return;

    const int hl = lane & 15;          // M index (A) / N index (B,D) within half
    const int kh = (lane >> 4) << 1;   // half-wave K offset

    v8f acc = {0.f, 0.f, 0.f, 0.f, 0.f, 0.f, 0.f, 0.f};
    #pragma unroll
    for (int k0 = 0; k0 < K; k0 += 4) {
        const int ka = k0 + kh;
        v2f a, b;
        a.x = As[hl * LDA + ka];
        a.y = As[hl * LDA + ka + 1];
        b.x = B[(size_t)ka * NCOLS + n0 + hl];
        b.y = B[(size_t)(ka + 1) * NCOLS + n0 + hl];
        acc = __builtin_amdgcn_wmma_f32_16x16x4_f32(
            false, a, false, b, (short)0, acc, false, false);
    }

    const int nc = n0 + hl;
    const int mb = mt * 16 + ((lane >> 4) << 3);   // rows 0..7 or 8..15
    #pragma unroll
    for (int v = 0; v < 8; ++v)
        C[(size_t)(mb + v) * NCOLS + nc] = acc[v];
}

// ============================================================
// el/er attention dots + init of segment-max / segment-sum
// ============================================================
template <int H, int D>
__global__ void attn_kernel(const float* __restrict__ feat,
                            const float* __restrict__ al,
                            const float* __restrict__ ar,
                            float* __restrict__ el, float* __restrict__ er,
                            unsigned* __restrict__ menc, float* __restrict__ s) {
    int i = blockIdx.x * blockDim.x + threadIdx.x;   // i = n*H + h
    if (i >= N_NODES * H) return;
    int h = i % H;
    const float* f = feat + (size_t)(i / H) * (H * D) + h * D;
    const float* alh = al + h * D;
    const float* arh = ar + h * D;
    float sl = 0.f, sr = 0.f;
    #pragma unroll 8
    for (int d = 0; d < D; ++d) {
        float fv = f[d];
        sl += fv * alh[d];
        sr += fv * arh[d];
    }
    el[i] = sl;
    er[i] = sr;
    menc[i] = 0u;     // encoded -inf
    s[i] = 0.f;
}

// ============================================================
// Edge pass 1: segment max via native u32 atomic max
// ============================================================
template <int H>
__global__ void edge_max_kernel(const int* __restrict__ src, const int* __restrict__ dst,
                                const float* __restrict__ el, const float* __restrict__ er,
                                unsigned* __restrict__ menc) {
    int i = blockIdx.x * blockDim.x + threadIdx.x;
    if (i >= E_EDGES * H) return;
    int e = i / H, h = i - e * H;
    float v = el[src[e] * H + h] + er[dst[e] * H + h];
    v = v >= 0.f ? v : SLOPE * v;
    atomicMax(&menc[dst[e] * H + h], fenc(v));
}

// ============================================================
// Edge pass 2: softmax denominator (recompute logit)
// ============================================================
template <int H>
__global__ void edge_sum_kernel(const int* __restrict__ src, const int* __restrict__ dst,
                                const float* __restrict__ el, const float* __restrict__ er,
                                const unsigned* __restrict__ menc, float* __restrict__ s) {
    int i = blockIdx.x * blockDim.x + threadIdx.x;
    if (i >= E_EDGES * H) return;
    int e = i / H, h = i - e * H;
    int d = dst[e];
    float v = el[src[e] * H + h] + er[d * H + h];
    v = v >= 0.f ? v : SLOPE * v;
    float ex = __expf(v - fdec(menc[d * H + h]));
    unsafeAtomicAdd(&s[d * H + h], ex);
}

__global__ void sinv_kernel(float* __restrict__ s, int n) {
    int i = blockIdx.x * blockDim.x + threadIdx.x;
    if (i >= n) return;
    float v = s[i];
    s[i] = (v == 0.f) ? 1.f : (1.f / v);
}

// ============================================================
// Edge pass 3: alpha-weighted gather/scatter (float4 gathers,
// native global_atomic_add_f32 scatters)
// ============================================================
template <int H, int D>
__global__ void edge_scatter_kernel(const int* __restrict__ src, const int* __restrict__ dst,
                                    const float* __restrict__ el, const float* __restrict__ er,
                                    const unsigned* __restrict__ menc,
                                    const float* __restrict__ sinv,
                                    const float* __restrict__ feat,
                                    float* __restrict__ acc) {
    int i = blockIdx.x * blockDim.x + threadIdx.x;
    if (i >= E_EDGES * H) return;
    int e = i / H, h = i - e * H;
    int sN = src[e], dN = dst[e];
    float v = el[sN * H + h] + er[dN * H + h];
    v = v >= 0.f ? v : SLOPE * v;
    float alpha = __expf(v - fdec(menc[dN * H + h])) * sinv[dN * H + h];

    const float4* f = (const float4*)(feat + (size_t)sN * (H * D) + h * D);
    float* a = acc + (size_t)dN * (H * D) + h * D;
    #pragma unroll
    for (int d4 = 0; d4 < D / 4; ++d4) {
        float4 fv = f[d4];
        unsafeAtomicAdd(a + d4 * 4 + 0, fv.x * alpha);
        unsafeAtomicAdd(a + d4 * 4 + 1, fv.y * alpha);
        unsafeAtomicAdd(a + d4 * 4 + 2, fv.z * alpha);
        unsafeAtomicAdd(a + d4 * 4 + 3, fv.w * alpha);
    }
}

// ============================================================
// elementwise helpers
// ============================================================
__global__ void zero_kernel(float* __restrict__ p, int n) {
    int i = blockIdx.x * blockDim.x + threadIdx.x;
    if (i < n) p[i] = 0.f;
}
__global__ void copy_kernel(float* __restrict__ d, const float* __restrict__ s, int n) {
    int i = blockIdx.x * blockDim.x + threadIdx.x;
    if (i < n) d[i] = s[i];
}
__global__ void elu_kernel(float* __restrict__ p, int n) {
    int i = blockIdx.x * blockDim.x + threadIdx.x;
    if (i < n) {
        float x = p[i];
        p[i] = x > 0.f ? x : expm1f(x);
    }
}
__global__ void mean_heads_kernel(const float* __restrict__ acc, float* __restrict__ out) {
    int i = blockIdx.x * blockDim.x + threadIdx.x;   // i = n*64 + c
    if (i >= N_NODES * 64) return;
    int n = i >> 6, c = i & 63;
    const float* a = acc + (size_t)n * 384 + c;
    float sum = 0.f;
    #pragma unroll
    for (int h = 0; h < 6; ++h) sum += a[h * 64];
    out[i] = sum * (1.f / 6.f);
}

// ============================================================
// launcher
// ============================================================
extern "C" void kernel_launch(void* const* d_in, const int* in_sizes, int n_in,
                              void* d_out, int out_size, void* d_ws, size_t ws_size,
                              hipStream_t stream) {
    const int*   src   = (const int*)d_in[0];
    const int*   dst   = (const int*)d_in[1];
    const float* embed = (const float*)d_in[2];
    const float* W0    = (const float*)d_in[3];
    const float* al0   = (const float*)d_in[4];
    const float* ar0   = (const float*)d_in[5];
    const float* W1    = (const float*)d_in[6];
    const float* al1   = (const float*)d_in[7];
    const float* ar1   = (const float*)d_in[8];
    const float* W2    = (const float*)d_in[9];
    const float* al2   = (const float*)d_in[10];
    const float* ar2   = (const float*)d_in[11];
    const float* Wres  = (const float*)d_in[12];
    float* out = (float*)d_out;

    // ---- workspace carve-up (all offsets 256B aligned) ----
    char* ws = (char*)d_ws;
    size_t off = 0;
    auto alloc = [&](size_t bytes) -> void* {
        void* p = ws + off;
        off += (bytes + 255) & ~(size_t)255;
        return p;
    };
    float*    el    = (float*)alloc((size_t)N_NODES * 6 * 4);
    float*    er    = (float*)alloc((size_t)N_NODES * 6 * 4);
    float*    sden  = (float*)alloc((size_t)N_NODES * 6 * 4);   // sum -> reciprocal in place
    unsigned* menc  = (unsigned*)alloc((size_t)N_NODES * 6 * 4);
    float*    h2buf = (float*)alloc((size_t)N_NODES * 128 * 4); // acc1 -> h2
    float*    bigA  = (float*)alloc((size_t)N_NODES * 384 * 4); // feat0/feat1 -> feat2
    float*    bigB  = (float*)alloc((size_t)N_NODES * 384 * 4); // acc0/h1 -> acc2

    auto cdiv = [](int a, int b) { return (a + b - 1) / b; };
    const int MT = N_NODES / 16;   // 3125, exact

    // ================= layer 0: feat0 = embed @ W0, H=4, D=32 =================
    wmma_gemm_kernel<64, 128><<<dim3(MT, 1), 256, 0, stream>>>(embed, W0, bigA);
    attn_kernel<4, 32><<<cdiv(N_NODES * 4, 256), 256, 0, stream>>>(bigA, al0, ar0, el, er, menc, sden);
    edge_max_kernel<4><<<cdiv(E_EDGES * 4, 256), 256, 0, stream>>>(src, dst, el, er, menc);
    edge_sum_kernel<4><<<cdiv(E_EDGES * 4, 256), 256, 0, stream>>>(src, dst, el, er, menc, sden);
    sinv_kernel<<<cdiv(N_NODES * 4, 256), 256, 0, stream>>>(sden, N_NODES * 4);
    zero_kernel<<<cdiv(N_NODES * 128, 256), 256, 0, stream>>>(bigB, N_NODES * 128);
    edge_scatter_kernel<4, 32><<<cdiv(E_EDGES * 4, 256), 256, 0, stream>>>(
        src, dst, el, er, menc, sden, bigA, bigB);
    elu_kernel<<<cdiv(N_NODES * 128, 256), 256, 0, stream>>>(bigB, N_NODES * 128);  // h1 in bigB

    // ================= layer 1: feat1 = h1 @ W1, identity residual =================
    wmma_gemm_kernel<128, 128><<<dim3(MT, 1), 256, 0, stream>>>(bigB, W1, bigA);
    attn_kernel<4, 32><<<cdiv(N_NODES * 4, 256), 256, 0, stream>>>(bigA, al1, ar1, el, er, menc, sden);
    edge_max_kernel<4><<<cdiv(E_EDGES * 4, 256), 256, 0, stream>>>(src, dst, el, er, menc);
    edge_sum_kernel<4><<<cdiv(E_EDGES * 4, 256), 256, 0, stream>>>(src, dst, el, er, menc, sden);
    sinv_kernel<<<cdiv(N_NODES * 4, 256), 256, 0, stream>>>(sden, N_NODES * 4);
    copy_kernel<<<cdiv(N_NODES * 128, 256), 256, 0, stream>>>(h2buf, bigB, N_NODES * 128); // res = h1
    edge_scatter_kernel<4, 32><<<cdiv(E_EDGES * 4, 256), 256, 0, stream>>>(
        src, dst, el, er, menc, sden, bigA, h2buf);
    elu_kernel<<<cdiv(N_NODES * 128, 256), 256, 0, stream>>>(h2buf, N_NODES * 128);  // h2

    // ================= layer 2: H=6, D=64, linear residual, head mean =================
    wmma_gemm_kernel<128, 384><<<dim3(MT, 3), 256, 0, stream>>>(h2buf, W2, bigA);   // feat2
    wmma_gemm_kernel<128, 384><<<dim3(MT, 3), 256, 0, stream>>>(h2buf, Wres, bigB); // acc2 = residual
    attn_kernel<6, 64><<<cdiv(N_NODES * 6, 256), 256, 0, stream>>>(bigA, al2, ar2, el, er, menc, sden);
    edge_max_kernel<6><<<cdiv(E_EDGES * 6, 256), 256, 0, stream>>>(src, dst, el, er, menc);
    edge_sum_kernel<6><<<cdiv(E_EDGES * 6, 256), 256, 0, stream>>>(src, dst, el, er, menc, sden);
    sinv_kernel<<<cdiv(N_NODES * 6, 256), 256, 0, stream>>>(sden, N_NODES * 6);
    edge_scatter_kernel<6, 64><<<cdiv(E_EDGES * 6, 256), 256, 0, stream>>>(
        src, dst, el, er, menc, sden, bigA, bigB);
    mean_heads_kernel<<<cdiv(N_NODES * 64, 256), 256, 0, stream>>>(bigB, out);
}